// SelfEncoder_88802743812415
// MI455X (gfx1250) — compile-verified
//
#include <hip/hip_runtime.h>
#include <hip/hip_bf16.h>

typedef __attribute__((ext_vector_type(2))) float v2f;
typedef __attribute__((ext_vector_type(8))) float v8f;

#define KNN 20
#define NPTS 2048
#define NB 16

// ---------------------------------------------------------------------------
// sq[b,n] = sum_d X[b,d,n]^2
template <int D>
__global__ __launch_bounds__(256)
void sqnorm_kernel(const float* __restrict__ X, float* __restrict__ SQ,
                   int B, int N) {
  int i = blockIdx.x * 256 + threadIdx.x;
  if (i >= B * N) return;
  int b = i / N, n = i % N;
  const float* xp = X + (size_t)b * D * N + n;
  float s = 0.f;
#pragma unroll
  for (int d = 0; d < D; ++d) {
    float v = xp[(size_t)d * N];
    s += v * v;
  }
  SQ[i] = s;
}

// ---------------------------------------------------------------------------
// Graph self-attention, fused: WMMA Gram tiles -> dist -> ballot-filtered
// running top-K -> softmax -> weighted neighbor-diff agg -> H=[x;agg] [B,2D,N]
// One workgroup = (batch b, 16 query points). 256 threads = 8 waves.
template <int D>
__global__ __launch_bounds__(256)
void gsa_kernel(const float* __restrict__ X, const float* __restrict__ SQ,
                float* __restrict__ H, int B, int N) {
  constexpr int KSTEPS = (D + 3) / 4;

  __shared__ float s_xq[64 * 16];         // [d][m], D<=64
  __shared__ float s_sqq[16];
  __shared__ float s_dist[16 * 128];      // 16 rows x 128 candidate cols
  __shared__ float s_topv[16 * KNN];
  __shared__ int   s_topi[16 * KNN];
  __shared__ float s_wgt[16 * KNN];

  const int b     = blockIdx.x;
  const int qbase = blockIdx.y * 16;
  const int tid   = threadIdx.x;
  const int lane  = tid & 31;
  const int wave  = tid >> 5;             // 0..7
  const int halfk = lane >> 4;            // 0/1
  const int ln    = lane & 15;

  // stage query features + norms into LDS
  for (int i = tid; i < D * 16; i += 256) {
    int d = i >> 4, m = i & 15;
    s_xq[i] = X[((size_t)b * D + d) * N + qbase + m];
  }
  if (tid < 16) s_sqq[tid] = SQ[b * N + qbase + tid];
  for (int i = tid; i < 16 * KNN; i += 256) {
    s_topv[i] = -3.402823466e38f;
    s_topi[i] = 0;
  }
  __syncthreads();

  // A fragments (16x4 f32 tiles of the query block) are chunk-invariant:
  // preload into registers. lanes0-15 hold K=k0,k0+1; lanes16-31 K=k0+2,k0+3.
  v2f afrag[KSTEPS];
#pragma unroll
  for (int ks = 0; ks < KSTEPS; ++ks) {
    int da0 = ks * 4 + 2 * halfk;
    afrag[ks].x = (da0 < D)     ? s_xq[da0 * 16 + ln]       : 0.f;
    afrag[ks].y = (da0 + 1 < D) ? s_xq[(da0 + 1) * 16 + ln] : 0.f;
  }

  for (int ch = 0; ch < N; ch += 128) {
    const int cb = ch + wave * 16;        // this wave's 16-column tile
    const float* xb = X + (size_t)b * D * N + cb + ln;  // per-lane column base
    v8f acc = {};
#pragma unroll
    for (int ks = 0; ks < KSTEPS; ++ks) {
      // B (4x16 f32): v0 rows K=k0(+halfk), v1 rows K=k0+2(+halfk), N=ln
      int db0 = ks * 4 + halfk, db1 = ks * 4 + 2 + halfk;
      v2f bf;
      bf.x = (db0 < D) ? xb[(size_t)db0 * N] : 0.f;
      bf.y = (db1 < D) ? xb[(size_t)db1 * N] : 0.f;
      acc = __builtin_amdgcn_wmma_f32_16x16x4_f32(
          false, afrag[ks], false, bf, (short)0, acc, false, false);
    }
    // Gram -> negative squared distance, deposit tile to LDS
    const float sqc = SQ[b * N + cb + ln];
#pragma unroll
    for (int r = 0; r < 8; ++r) {
      int m = r + 8 * halfk;              // C/D layout: M = vgpr + 8*(lane>=16)
      s_dist[m * 128 + wave * 16 + ln] = 2.f * acc[r] - s_sqq[m] - sqc;
    }
    __syncthreads();

    // overlap: prefetch next chunk's candidate rows into the WGP-near cache
    // (locality 3 -> WGP scope: pull into all cache levels) while selection
    // runs on this chunk's tile.
    if (ch + 128 < N) {
      for (int d = wave; d < D; d += 8)
        __builtin_prefetch(&X[((size_t)b * D + d) * N + ch + 128 + lane * 4],
                           0, 3);
    }

    // running top-K: 2 rows per wave; ballot filters 32 candidates at a time
    // against the current list minimum (which only increases, so a stale
    // threshold admits a superset; lane 0 re-checks authoritatively).
    // Ballot bits processed low->high keeps ascending candidate index, so
    // ties resolve to the lower index like lax.top_k.
#pragma unroll
    for (int rr = 0; rr < 2; ++rr) {
      const int r = wave * 2 + rr;
      float* tv = &s_topv[r * KNN];
      int*   ti = &s_topi[r * KNN];
      for (int c0 = 0; c0 < 128; c0 += 32) {
        float v   = s_dist[r * 128 + c0 + lane];
        float thr = tv[KNN - 1];
        unsigned mask = (unsigned)__ballot(v > thr);
        while (mask) {
          int s = __builtin_ctz(mask);
          mask &= mask - 1;
          float vs = __shfl(v, s);
          if (lane == 0 && vs > tv[KNN - 1]) {
            int p = KNN - 1;
            while (p > 0 && tv[p - 1] < vs) {
              tv[p] = tv[p - 1]; ti[p] = ti[p - 1]; --p;
            }
            tv[p] = vs; ti[p] = ch + c0 + s;
          }
        }
      }
    }
    __syncthreads();
  }

  // softmax over the K kept (descending-sorted) distances
  if (tid < 16) {
    float mx = s_topv[tid * KNN];
    float ssum = 0.f;
    for (int k = 0; k < KNN; ++k) {
      float e = __expf(s_topv[tid * KNN + k] - mx);
      s_wgt[tid * KNN + k] = e;
      ssum += e;
    }
    float inv = 1.f / ssum;
    for (int k = 0; k < KNN; ++k) s_wgt[tid * KNN + k] *= inv;
  }
  __syncthreads();

  // aggregation: agg[t,d] = sum_k w * (X[b,d,idx_k] - xq[d,t]); H = [x ; agg]
  const int t = tid & 15;
  const size_t n = qbase + t;
  for (int d = tid >> 4; d < D; d += 16) {
    float xq = s_xq[d * 16 + t];
    float agg = 0.f;
#pragma unroll
    for (int k = 0; k < KNN; ++k) {
      int idx = s_topi[t * KNN + k];
      agg += s_wgt[t * KNN + k] * (X[((size_t)b * D + d) * N + idx] - xq);
    }
    H[((size_t)b * (2 * D) + d) * N + n]     = xq;
    H[((size_t)b * (2 * D) + D + d) * N + n] = agg;
  }
}

// ---------------------------------------------------------------------------
// Y[b,o,n] = sum_c W[o,c] * H[b,c,n]   (WMMA GEMM, 16 chans x 128 pts / WG)
template <int C>
__global__ __launch_bounds__(256)
void conv_kernel(const float* __restrict__ W, const float* __restrict__ H,
                 float* __restrict__ Y, int B, int O, int N) {
  constexpr int KSTEPS = (C + 3) / 4;
  __shared__ float s_w[16 * 128];         // [m][c], C <= 128
  const int nct   = N / 128;
  const int b     = blockIdx.x / nct;
  const int ct    = blockIdx.x % nct;
  const int obase = blockIdx.y * 16;
  const int tid   = threadIdx.x;
  const int lane  = tid & 31;
  const int wave  = tid >> 5;
  const int halfk = lane >> 4;
  const int ln    = lane & 15;
  const int cb    = ct * 128 + wave * 16;

  for (int i = tid; i < 16 * C; i += 256) {
    int m = i / C, c = i % C;
    s_w[i] = W[(obase + m) * C + c];
  }
  __syncthreads();

  const float* hb = H + (size_t)b * C * N + cb + ln;
  v8f acc = {};
#pragma unroll
  for (int ks = 0; ks < KSTEPS; ++ks) {
    int da0 = ks * 4 + 2 * halfk;
    v2f a;
    a.x = (da0 < C)     ? s_w[ln * C + da0]     : 0.f;
    a.y = (da0 + 1 < C) ? s_w[ln * C + da0 + 1] : 0.f;
    int db0 = ks * 4 + halfk, db1 = ks * 4 + 2 + halfk;
    v2f bf;
    bf.x = (db0 < C) ? hb[(size_t)db0 * N] : 0.f;
    bf.y = (db1 < C) ? hb[(size_t)db1 * N] : 0.f;
    acc = __builtin_amdgcn_wmma_f32_16x16x4_f32(
        false, a, false, bf, (short)0, acc, false, false);
  }
#pragma unroll
  for (int r = 0; r < 8; ++r) {
    int m = r + 8 * halfk;
    Y[((size_t)b * O + obase + m) * N + cb + ln] = acc[r];
  }
}

// ---------------------------------------------------------------------------
// Per-channel batch stats (deterministic fixed-order tree; no atomics).
// mr[o] = mean, mr[O+o] = rsqrt(var + eps)
__global__ __launch_bounds__(256)
void stats_kernel(const float* __restrict__ Y, float* __restrict__ mr,
                  int B, int O, int N) {
  __shared__ float rs[256], rs2[256];
  const int o = blockIdx.x;
  const int tid = threadIdx.x;
  const int total = B * N;
  float s = 0.f, s2 = 0.f;
  for (int i = tid; i < total; i += 256) {
    int b = i / N, n = i % N;
    float v = Y[((size_t)b * O + o) * N + n];
    s += v; s2 += v * v;
  }
  rs[tid] = s; rs2[tid] = s2;
  __syncthreads();
  for (int st = 128; st > 0; st >>= 1) {
    if (tid < st) { rs[tid] += rs[tid + st]; rs2[tid] += rs2[tid + st]; }
    __syncthreads();
  }
  if (tid == 0) {
    float mean = rs[0] / total;
    float var  = rs2[0] / total - mean * mean;
    mr[o]     = mean;
    mr[O + o] = rsqrtf(var + 1e-5f);
  }
}

// ---------------------------------------------------------------------------
// BN apply + LeakyReLU(0.2)
__global__ __launch_bounds__(256)
void apply_kernel(const float* __restrict__ Y, const float* __restrict__ mr,
                  const float* __restrict__ g, const float* __restrict__ be,
                  float* __restrict__ OUT, int B, int O, int N) {
  size_t i = (size_t)blockIdx.x * 256 + threadIdx.x;
  size_t tot = (size_t)B * O * N;
  if (i >= tot) return;
  int o = (int)((i / N) % O);
  float v = (Y[i] - mr[o]) * mr[O + o] * g[o] + be[o];
  OUT[i] = v >= 0.f ? v : 0.2f * v;
}

// ---------------------------------------------------------------------------
extern "C" void kernel_launch(void* const* d_in, const int* in_sizes, int n_in,
                              void* d_out, int out_size, void* d_ws, size_t ws_size,
                              hipStream_t stream) {
  (void)in_sizes; (void)n_in; (void)out_size; (void)ws_size;
  const float* x   = (const float*)d_in[0];
  const float* W1  = (const float*)d_in[1];
  const float* W2  = (const float*)d_in[2];
  const float* W3  = (const float*)d_in[3];
  const float* g1p = (const float*)d_in[4];
  const float* b1p = (const float*)d_in[5];
  const float* g2p = (const float*)d_in[6];
  const float* b2p = (const float*)d_in[7];
  const float* g3p = (const float*)d_in[8];
  const float* b3p = (const float*)d_in[9];

  const int B = NB, N = NPTS;
  float* out = (float*)d_out;
  float* x1 = out;                                   // [16,64,2048]
  float* x2 = out + (size_t)B * 64 * N;              // [16,64,2048]
  float* x3 = out + (size_t)2 * B * 64 * N;          // [16,128,2048]

  float* h  = (float*)d_ws;                          // up to [16,128,2048]
  float* y  = h  + (size_t)B * 128 * N;              // up to [16,128,2048]
  float* sq = y  + (size_t)B * 128 * N;              // [16,2048]
  float* mr = sq + (size_t)B * N;                    // [2*128]

  dim3 blk(256);
  dim3 gg(B, N / 16);

  // ---- stage 1: D=3, C=6, O=64 ----
  sqnorm_kernel<3><<<(B * N + 255) / 256, blk, 0, stream>>>(x, sq, B, N);
  gsa_kernel<3><<<gg, blk, 0, stream>>>(x, sq, h, B, N);
  conv_kernel<6><<<dim3(B * (N / 128), 64 / 16), blk, 0, stream>>>(
      W1, h, y, B, 64, N);
  stats_kernel<<<64, blk, 0, stream>>>(y, mr, B, 64, N);
  apply_kernel<<<(unsigned)(((size_t)B * 64 * N + 255) / 256), blk, 0, stream>>>(
      y, mr, g1p, b1p, x1, B, 64, N);

  // ---- stage 2: D=64, C=128, O=64 ----
  sqnorm_kernel<64><<<(B * N + 255) / 256, blk, 0, stream>>>(x1, sq, B, N);
  gsa_kernel<64><<<gg, blk, 0, stream>>>(x1, sq, h, B, N);
  conv_kernel<128><<<dim3(B * (N / 128), 64 / 16), blk, 0, stream>>>(
      W2, h, y, B, 64, N);
  stats_kernel<<<64, blk, 0, stream>>>(y, mr, B, 64, N);
  apply_kernel<<<(unsigned)(((size_t)B * 64 * N + 255) / 256), blk, 0, stream>>>(
      y, mr, g2p, b2p, x2, B, 64, N);

  // ---- stage 3: D=64, C=128, O=128 ----
  sqnorm_kernel<64><<<(B * N + 255) / 256, blk, 0, stream>>>(x2, sq, B, N);
  gsa_kernel<64><<<gg, blk, 0, stream>>>(x2, sq, h, B, N);
  conv_kernel<128><<<dim3(B * (N / 128), 128 / 16), blk, 0, stream>>>(
      W3, h, y, B, 128, N);
  stats_kernel<<<128, blk, 0, stream>>>(y, mr, B, 128, N);
  apply_kernel<<<(unsigned)(((size_t)B * 128 * N + 255) / 256), blk, 0, stream>>>(
      y, mr, g3p, b3p, x3, B, 128, N);
}